// MemoryEfficientContrastiveLossForSeq_9947144257730
// MI455X (gfx1250) — compile-verified
//
#include <hip/hip_runtime.h>
#include <hip/hip_bf16.h>

// Problem constants from the reference: preds/targets/masks are (B=256, L=512, N=4).
// Row-major flat index: (b*L + l)*N + n = b*2048 + (l*4 + n). One block per column
// c = l*4+n; gridDim.x = 2048.
#define BATCH 256
#define COLS  2048      // L*N
#define COLSTRIDE 2048  // L*N elements between consecutive b for fixed (l,n)

typedef __attribute__((ext_vector_type(2))) float v2f;
typedef __attribute__((ext_vector_type(8))) float v8f;

#define LOG2E 1.44269504088896340736f
#define LN2   0.69314718055994530942f

// ---------------------------------------------------------------------------
// Kernel 1: one 256-thread block (8 wave32 waves) per (l,n) column.
// Stage the 256-long column of preds/targets/masks in LDS, then cover the
// 16x16 tile grid of batch pairs. Only tiles with j0 >= i0 are processed
// (136 of 256); the strict triangle j>i is enforced per element.
//
// Per tile, three V_WMMA_F32_16X16X4_F32 ops build:
//   Dp[i][j] = p_i - p_j      (A = [p,1,0,0], B rows = [1; -p; 0; 0])
//   Dt[i][j] = t_i - t_j      (same construction with targets)
//   Dm[i][j] = m_i * m_j      (K=1 outer product of mask floats)
// then each lane evaluates BCE-with-logits on its 8 C-layout elements using
// raw v_exp_f32/v_log_f32 (exp2/log2) — argument ranges are provably safe
// (exp arg <= 0, log arg in [1,2]), so no denorm/overflow fixups needed.
// C/D layout (ISA 7.12.2): VGPR r, lanes 0-15 -> M=r,   N=lane
//                                  lanes 16-31 -> M=r+8, N=lane-16
// ---------------------------------------------------------------------------
__global__ __launch_bounds__(256)
void pairwise_bce_wmma_kernel(const float* __restrict__ preds,
                              const float* __restrict__ targets,
                              const unsigned char* __restrict__ masks,
                              float* __restrict__ wsLoss,
                              float* __restrict__ wsCnt) {
    __shared__ float sp[BATCH];
    __shared__ float st[BATCH];
    __shared__ float sm[BATCH];

    const int c   = blockIdx.x;      // column = l*4 + n
    const int tid = threadIdx.x;     // 0..255 -> b

    // Hint the column into cache (global_prefetch_b8), then stage to LDS.
    __builtin_prefetch(&preds[(size_t)tid * COLSTRIDE + c], 0, 1);
    sp[tid] = preds  [(size_t)tid * COLSTRIDE + c];
    st[tid] = targets[(size_t)tid * COLSTRIDE + c];
    sm[tid] = masks  [(size_t)tid * COLSTRIDE + c] ? 1.0f : 0.0f;
    __syncthreads();

    const int lane = tid & 31;
    const int wave = tid >> 5;       // 0..7
    const int lm   = lane & 15;
    const bool lo  = (lane < 16);

    float accL = 0.0f;   // sum of masked BCE
    float accC = 0.0f;   // count of valid pairs

    // 136 upper-triangular tiles split evenly: wave w takes tt = w*17 + k.
    // Decode the first (ti,tj) once, then walk the triangle incrementally.
    int ti = 0, tj;
    {
        int rem = wave * 17;
        while (rem >= 16 - ti) { rem -= 16 - ti; ++ti; }
        tj = ti + rem;
    }

    for (int k = 0; k < 17; ++k) {
        const int i0 = ti << 4;
        const int j0 = tj << 4;

        // --- build WMMA fragments (selects lower via v_cndmask, EXEC stays full)
        v2f ap, bp, at, bt, am, bm;
        ap.x = lo ? sp[i0 + lm] : 0.0f;  ap.y = lo ? 1.0f : 0.0f;
        bp.x = lo ? 1.0f        : 0.0f;  bp.y = lo ? -sp[j0 + lm] : 0.0f;
        at.x = lo ? st[i0 + lm] : 0.0f;  at.y = lo ? 1.0f : 0.0f;
        bt.x = lo ? 1.0f        : 0.0f;  bt.y = lo ? -st[j0 + lm] : 0.0f;
        am.x = lo ? sm[i0 + lm] : 0.0f;  am.y = 0.0f;
        bm.x = lo ? sm[j0 + lm] : 0.0f;  bm.y = 0.0f;

        v8f cz = {};
        // 8 args: (neg_a, A, neg_b, B, c_mod, C, reuse_a, reuse_b)
        v8f dp = __builtin_amdgcn_wmma_f32_16x16x4_f32(false, ap, false, bp,
                                                       (short)0, cz, false, false);
        v8f dt = __builtin_amdgcn_wmma_f32_16x16x4_f32(false, at, false, bt,
                                                       (short)0, cz, false, false);
        v8f dm = __builtin_amdgcn_wmma_f32_16x16x4_f32(false, am, false, bm,
                                                       (short)0, cz, false, false);

        const int mbase = i0 + (lo ? 0 : 8);
        const int j     = j0 + lm;
#pragma unroll
        for (int r = 0; r < 8; ++r) {
            const int i = mbase + r;
            float x  = dp[r];
            float tg = (dt[r] > 1e-3f) ? 1.0f : 0.0f;
            float sel = (j > i) ? dm[r] : 0.0f;       // 0/1 pair mask * triangle
            // softplus(-|x|) = ln2 * log2(1 + 2^(-|x|*log2e)); arg in [1,2]
            float e  = __builtin_amdgcn_exp2f(-fabsf(x) * LOG2E);
            float l2 = __builtin_amdgcn_logf(1.0f + e);
            // bce = max(x,0) - x*tg + ln2*l2
            float bce = fmaf(LN2, l2, fmaf(-x, tg, fmaxf(x, 0.0f)));
            accL = fmaf(sel, bce, accL);
            accC += sel;
        }

        // advance (ti,tj) along the upper triangle
        ++tj;
        if (tj == 16) { ++ti; tj = ti; }
    }

    // Deterministic block reduction (reuse sp/st after a barrier).
    __syncthreads();
    sp[tid] = accL;
    st[tid] = accC;
    __syncthreads();
#pragma unroll
    for (int s = 128; s > 0; s >>= 1) {
        if (tid < s) { sp[tid] += sp[tid + s]; st[tid] += st[tid + s]; }
        __syncthreads();
    }
    if (tid == 0) {
        wsLoss[c] = sp[0];
        wsCnt [c] = st[0];
    }
}

// ---------------------------------------------------------------------------
// Kernel 2: fixed-order reduction of 2048 column partials into 4 per-feature
// sums (reduction stride stops at 4, so n = c%4 classes never mix), then the
// per-feature normalization and mean. Fully deterministic.
// ---------------------------------------------------------------------------
__global__ __launch_bounds__(256)
void finalize_kernel(const float* __restrict__ wsLoss,
                     const float* __restrict__ wsCnt,
                     float* __restrict__ out) {
    __shared__ float shL[256];
    __shared__ float shC[256];
    const int t = threadIdx.x;
    float aL = 0.0f, aC = 0.0f;
    for (int c = t; c < COLS; c += 256) { aL += wsLoss[c]; aC += wsCnt[c]; }
    shL[t] = aL; shC[t] = aC;
    __syncthreads();
#pragma unroll
    for (int s = 128; s >= 4; s >>= 1) {
        if (t < s) { shL[t] += shL[t + s]; shC[t] += shC[t + s]; }
        __syncthreads();
    }
    if (t == 0) {
        float total = 0.0f;
#pragma unroll
        for (int n = 0; n < 4; ++n) {
            float ls = shL[n];
            float ms = shC[n];
            total += (ms > 0.0f) ? ls / (ms + 1e-8f) : 0.0f;
        }
        out[0] = total * 0.25f;   // divide by N=4
    }
}

extern "C" void kernel_launch(void* const* d_in, const int* in_sizes, int n_in,
                              void* d_out, int out_size, void* d_ws, size_t ws_size,
                              hipStream_t stream) {
    const float*         preds   = (const float*)d_in[0];
    const float*         targets = (const float*)d_in[1];
    const unsigned char* masks   = (const unsigned char*)d_in[2];  // jnp.bool_ = 1 byte

    float* wsLoss = (float*)d_ws;          // 2048 floats
    float* wsCnt  = wsLoss + COLS;         // 2048 floats (16 KB total)
    float* out    = (float*)d_out;

    pairwise_bce_wmma_kernel<<<COLS, 256, 0, stream>>>(preds, targets, masks,
                                                       wsLoss, wsCnt);
    finalize_kernel<<<1, 256, 0, stream>>>(wsLoss, wsCnt, out);
}